// GraphAttentionHead_78546361909517
// MI455X (gfx1250) — compile-verified
//
#include <hip/hip_runtime.h>
#include <hip/hip_bf16.h>
#include <math.h>

// GAT head, N=8192, IN=512, OUT=256.
// Pipeline: hw = h@w (f32 WMMA) -> transpose hw -> e = hw@a -> masked row
// softmax stats -> fused att@hw (f32 WMMA, att generated in-register).

#define N_NODES 8192
#define IN_F    512
#define OUT_F   256
#define SLOPE   0.2f

typedef __attribute__((ext_vector_type(2))) float v2f;
typedef __attribute__((ext_vector_type(8))) float v8f;

// ---------------------------------------------------------------------------
// K0: wT[n][k] = w[k][n]  (512x256 -> 256x512, tiny)
// ---------------------------------------------------------------------------
__global__ void k_transpose_w(const float* __restrict__ w, float* __restrict__ wT) {
  const int idx = blockIdx.x * blockDim.x + threadIdx.x;  // k*256 + n
  if (idx >= IN_F * OUT_F) return;
  const int k = idx >> 8;
  const int n = idx & 255;
  wT[(size_t)n * IN_F + k] = w[idx];
}

// ---------------------------------------------------------------------------
// K1: hw = h @ w   [8192x512]*[512x256], fp32 WMMA 16x16x4.
// Block = 128 thr (4 waves); block owns 16 rows x 256 cols; wave owns 16x64.
// B fragment pairs (K, K+1) are contiguous in wT -> single b64 load.
// ---------------------------------------------------------------------------
__global__ __launch_bounds__(128) void k_gemm_hw(const float* __restrict__ h,
                                                 const float* __restrict__ wT,
                                                 float* __restrict__ hw) {
  const int wave  = threadIdx.x >> 5;
  const int lane  = threadIdx.x & 31;
  const int r0    = blockIdx.x * 16;
  const int n0    = wave * 64;
  const int lrow  = lane & 15;          // M (A/C) or N (B/C) sub-index
  const int khalf = (lane >> 4) * 2;    // lanes 0-15 -> K0/K1, 16-31 -> K2/K3

  v8f acc[4] = {};
  const float* hrow = h + (size_t)(r0 + lrow) * IN_F;

  for (int k0 = 0; k0 < IN_F; k0 += 4) {
    const v2f a = *(const v2f*)(hrow + k0 + khalf);
#pragma unroll
    for (int t = 0; t < 4; ++t) {
      const int nb = n0 + t * 16 + lrow;
      const v2f b = *(const v2f*)(wT + (size_t)nb * IN_F + k0 + khalf);
      acc[t] = __builtin_amdgcn_wmma_f32_16x16x4_f32(false, a, false, b,
                                                     (short)0, acc[t], false, false);
    }
  }
#pragma unroll
  for (int t = 0; t < 4; ++t)
#pragma unroll
    for (int i = 0; i < 8; ++i) {
      const int row = r0 + i + 8 * (lane >> 4);
      hw[(size_t)row * OUT_F + n0 + t * 16 + lrow] = acc[t][i];
    }
}

// ---------------------------------------------------------------------------
// K1b: hwT[n][j] = hw[j][n]  (8192x256 -> 256x8192), LDS 32x32 tile transpose,
// coalesced on both sides.
// ---------------------------------------------------------------------------
__global__ __launch_bounds__(256) void k_transpose_hw(const float* __restrict__ hw,
                                                      float* __restrict__ hwT) {
  __shared__ float tile[32][33];
  const int tx = threadIdx.x & 31;       // fast dim
  const int ty = threadIdx.x >> 5;       // 0..7
  const int n0 = blockIdx.x * 32;        // column (OUT_F) tile base
  const int j0 = blockIdx.y * 32;        // row (node) tile base

#pragma unroll
  for (int i = 0; i < 32; i += 8)
    tile[ty + i][tx] = hw[(size_t)(j0 + ty + i) * OUT_F + n0 + tx];
  __syncthreads();
#pragma unroll
  for (int i = 0; i < 32; i += 8)
    hwT[(size_t)(n0 + ty + i) * N_NODES + j0 + tx] = tile[tx][ty + i];
}

// ---------------------------------------------------------------------------
// K2: e = hw @ a -> e1, e2. One thread per row; hwT column reads are coalesced.
// ---------------------------------------------------------------------------
__global__ void k_e(const float* __restrict__ hwT, const float* __restrict__ a,
                    float* __restrict__ e1, float* __restrict__ e2) {
  const int i = blockIdx.x * blockDim.x + threadIdx.x;
  if (i >= N_NODES) return;
  float s0 = 0.f, s1 = 0.f;
  for (int c = 0; c < OUT_F; ++c) {
    const float v = hwT[(size_t)c * N_NODES + i];
    s0 = fmaf(v, a[2 * c + 0], s0);
    s1 = fmaf(v, a[2 * c + 1], s1);
  }
  e1[i] = s0;
  e2[i] = s1;
}

// ---------------------------------------------------------------------------
// K3: per-row softmax stats over masked logits. One wave per row; each lane
// loads 4 adjacency bytes per u32 (full 128B line per wave), online (max,sum),
// then shuffle-tree merge.
// ---------------------------------------------------------------------------
__global__ __launch_bounds__(128) void k_stats(const unsigned char* __restrict__ adj,
                                               const float* __restrict__ e1,
                                               const float* __restrict__ e2,
                                               float* __restrict__ mrow,
                                               float* __restrict__ rlrow) {
  const int wave = threadIdx.x >> 5;
  const int lane = threadIdx.x & 31;
  const int row  = blockIdx.x * 4 + wave;
  const float a1 = e1[row];
  const unsigned char* arow = adj + (size_t)row * N_NODES;

  float m = -3.4e38f, l = 0.f;
  for (int j0 = lane * 4; j0 < N_NODES; j0 += 128) {
    const unsigned int wbits = *(const unsigned int*)(arow + j0);
#pragma unroll
    for (int t = 0; t < 4; ++t) {
      if ((wbits >> (8 * t)) & 0xffu) {
        float s = a1 + e2[j0 + t];
        s = (s > 0.f) ? s : SLOPE * s;
        if (s > m) {
          l = l * __expf(m - s) + 1.f;
          m = s;
        } else {
          l += __expf(s - m);
        }
      }
    }
  }
  for (int off = 16; off > 0; off >>= 1) {
    const float mo = __shfl_xor(m, off, 32);
    const float lo = __shfl_xor(l, off, 32);
    const float mn = fmaxf(m, mo);
    l = l * __expf(m - mn) + lo * __expf(mo - mn);
    m = mn;
  }
  if (lane == 0) {
    mrow[row]  = m;
    rlrow[row] = 1.f / l;   // self-loop guarantees l >= 1
  }
}

// ---------------------------------------------------------------------------
// K4: out = att @ hw, att generated on the fly into the WMMA A fragment:
//   p_ij = adj_ij ? exp(lrelu(e1_i + e2_j) - m_i) / l_i : 0
// Block = 4 waves; block owns 16 rows x 256 cols; wave owns 16x64 (4 C tiles).
// B fragments come from hwT: K-consecutive pair = one b64 load.
// ---------------------------------------------------------------------------
__global__ __launch_bounds__(128) void k_att_av(const unsigned char* __restrict__ adj,
                                                const float* __restrict__ hwT,
                                                const float* __restrict__ e1,
                                                const float* __restrict__ e2,
                                                const float* __restrict__ mrow,
                                                const float* __restrict__ rlrow,
                                                float* __restrict__ out) {
  const int wave  = threadIdx.x >> 5;
  const int lane  = threadIdx.x & 31;
  const int r0    = blockIdx.x * 16;
  const int n0    = wave * 64;
  const int lrow  = lane & 15;
  const int khalf = (lane >> 4) * 2;

  const int   row = r0 + lrow;
  const float a1  = e1[row];
  const float m   = mrow[row];
  const float rl  = rlrow[row];
  const unsigned char* arow = adj + (size_t)row * N_NODES;

  v8f acc[4] = {};

  for (int j0 = 0; j0 < N_NODES; j0 += 4) {
    const int jb = j0 + khalf;                               // even -> aligned
    const unsigned short ad = *(const unsigned short*)(arow + jb);
    const v2f e2p = *(const v2f*)(e2 + jb);
    float s0 = a1 + e2p.x;
    float s1 = a1 + e2p.y;
    s0 = (s0 > 0.f) ? s0 : SLOPE * s0;
    s1 = (s1 > 0.f) ? s1 : SLOPE * s1;
    v2f a;                                     // per-lane selects, EXEC stays full
    a.x = (ad & 0x00ffu) ? __expf(s0 - m) * rl : 0.f;
    a.y = (ad & 0xff00u) ? __expf(s1 - m) * rl : 0.f;
#pragma unroll
    for (int t = 0; t < 4; ++t) {
      const int nb = n0 + t * 16 + lrow;
      const v2f b = *(const v2f*)(hwT + (size_t)nb * N_NODES + jb);  // one b64
      acc[t] = __builtin_amdgcn_wmma_f32_16x16x4_f32(false, a, false, b,
                                                     (short)0, acc[t], false, false);
    }
  }
#pragma unroll
  for (int t = 0; t < 4; ++t)
#pragma unroll
    for (int i = 0; i < 8; ++i) {
      const int orow = r0 + i + 8 * (lane >> 4);
      out[(size_t)orow * OUT_F + n0 + t * 16 + lrow] = acc[t][i];
    }
}

// ---------------------------------------------------------------------------
extern "C" void kernel_launch(void* const* d_in, const int* in_sizes, int n_in,
                              void* d_out, int out_size, void* d_ws, size_t ws_size,
                              hipStream_t stream) {
  const float*         h   = (const float*)d_in[0];
  const unsigned char* adj = (const unsigned char*)d_in[1];  // jnp bool = 1 byte
  const float*         w   = (const float*)d_in[2];
  const float*         a   = (const float*)d_in[3];
  float*               out = (float*)d_out;

  float* ws   = (float*)d_ws;
  float* hw   = ws;                                    // 8192*256
  float* hwT  = hw  + (size_t)N_NODES * OUT_F;         // 256*8192
  float* wT   = hwT + (size_t)N_NODES * OUT_F;         // 256*512
  float* e1   = wT  + (size_t)OUT_F * IN_F;            // 8192
  float* e2   = e1 + N_NODES;                          // 8192
  float* mrow = e2 + N_NODES;                          // 8192
  float* rlw  = mrow + N_NODES;                        // 8192

  // 0) wT = w^T
  k_transpose_w<<<(IN_F * OUT_F) / 256, 256, 0, stream>>>(w, wT);
  // 1) hw = h @ w
  k_gemm_hw<<<N_NODES / 16, 128, 0, stream>>>(h, wT, hw);
  // 1b) hwT = hw^T
  {
    dim3 g(OUT_F / 32, N_NODES / 32);
    k_transpose_hw<<<g, 256, 0, stream>>>(hw, hwT);
  }
  // 2) e = hw @ a
  k_e<<<N_NODES / 256, 256, 0, stream>>>(hwT, a, e1, e2);
  // 3) row-wise masked softmax stats (max, 1/sum-exp)
  k_stats<<<N_NODES / 4, 128, 0, stream>>>(adj, e1, e2, mrow, rlw);
  // 4) out = softmax(mask(lrelu(e1_i+e2_j))) @ hw, fused
  k_att_av<<<N_NODES / 16, 128, 0, stream>>>(adj, hwT, e1, e2, mrow, rlw, out);
}